// StackMemory_70841190581009
// MI455X (gfx1250) — compile-verified
//
#include <hip/hip_runtime.h>
#include <math.h>

#define NB 4
#define SS 2048
#define HH 1024
#define NHEAD 16
#define NSLOTS 16
#define SDIM 32
#define HDIM 64
#define TOK_PER_WG 16
#define THREADS 512  // 16 wave32, one wave per head

typedef __attribute__((ext_vector_type(2))) float v2f;
typedef __attribute__((ext_vector_type(8))) float v8f;

// LDS layout (floats):
//  hs_s   : [16][1025]      (padded row -> conflict-free column reads)
//  log_s  : [16][48]
//  act_s  : [16][16][3]
//  k_s    : [16 heads][16 tok][33]
//  mem_s  : [16 heads][16 tok][33]
#define HS_STRIDE 1025
#define OFF_LOG   (16 * HS_STRIDE)
#define OFF_ACT   (OFF_LOG + 16 * 48)
#define OFF_K     (OFF_ACT + 16 * 48)
#define OFF_MEM   (OFF_K + 16 * 16 * 33)
#define SMEM_FLOATS (OFF_MEM + 16 * 16 * 33)

__global__ void __launch_bounds__(THREADS)
stackmem_kernel(const float* __restrict__ hs_g, const float* __restrict__ stack_g,
                const float* __restrict__ mask_g,
                const float* __restrict__ Wa, const float* __restrict__ ba,
                const float* __restrict__ Wd, const float* __restrict__ bd,
                const float* __restrict__ Wu, const float* __restrict__ bu,
                const float* __restrict__ Wg, const float* __restrict__ bg,
                const float* __restrict__ resw,
                float* __restrict__ out_g, float* __restrict__ ostack_g,
                float* __restrict__ omask_g)
{
    extern __shared__ float smem[];
    float* hs_s  = smem;
    float* log_s = smem + OFF_LOG;
    float* act_s = smem + OFF_ACT;
    float* k_s   = smem + OFF_K;
    float* mem_s = smem + OFF_MEM;

    const int tid  = threadIdx.x;
    const int lane = tid & 31;
    const int wave = tid >> 5;               // == head index
    const int mrow = lane & 15;              // M row / N col within WMMA half
    const int khi  = (lane >> 4) << 1;       // 0 for lanes 0-15, 2 for 16-31
    const int thalf = (lane >> 4) << 3;      // D-layout: M = r + 8*(lane>=16)
    const long long tok0 = (long long)blockIdx.x * TOK_PER_WG;

    // ---- Phase 1: stage the 16x1024 hidden tile into LDS (coalesced) ----
    {
        const float* src = hs_g + tok0 * HH;
        for (int i = tid; i < TOK_PER_WG * HH; i += THREADS) {
            int t = i >> 10;
            int c = i & 1023;
            hs_s[t * HS_STRIDE + c] = src[i];
        }
    }
    __syncthreads();

    // ---- Phase 2a: per-head k projection  k[16x32] = hs_head[16x64] @ Wd ----
    {
        const int h = wave;
        const float* arow = hs_s + mrow * HS_STRIDE + h * HDIM;
        v8f c0 = {}, c1 = {};
        #pragma unroll
        for (int k0 = 0; k0 < HDIM; k0 += 4) {
            v2f a; a.x = arow[k0 + khi]; a.y = arow[k0 + khi + 1];
            v2f b0, b1;
            b0.x = Wd[(k0 + khi) * SDIM + mrow];
            b0.y = Wd[(k0 + khi + 1) * SDIM + mrow];
            b1.x = Wd[(k0 + khi) * SDIM + 16 + mrow];
            b1.y = Wd[(k0 + khi + 1) * SDIM + 16 + mrow];
            c0 = __builtin_amdgcn_wmma_f32_16x16x4_f32(false, a, false, b0,
                                                       (short)0, c0, false, false);
            c1 = __builtin_amdgcn_wmma_f32_16x16x4_f32(false, a, false, b1,
                                                       (short)0, c1, false, false);
        }
        #pragma unroll
        for (int r = 0; r < 8; r++) {
            int t = r + thalf;
            k_s[(h * 16 + t) * 33 + mrow]      = c0[r] + bd[mrow];
            k_s[(h * 16 + t) * 33 + 16 + mrow] = c1[r] + bd[16 + mrow];
        }
    }

    // ---- Phase 2b: logits[16x48] = hs[16x1024] @ Wa  (waves 0..2, one N-tile each)
    if (wave < 3) {
        const int nt = wave;
        const float* arow = hs_s + mrow * HS_STRIDE;
        v8f c = {};
        for (int k0 = 0; k0 < HH; k0 += 4) {
            v2f a; a.x = arow[k0 + khi]; a.y = arow[k0 + khi + 1];
            v2f b;
            b.x = Wa[(k0 + khi) * 48 + nt * 16 + mrow];
            b.y = Wa[(k0 + khi + 1) * 48 + nt * 16 + mrow];
            c = __builtin_amdgcn_wmma_f32_16x16x4_f32(false, a, false, b,
                                                      (short)0, c, false, false);
        }
        #pragma unroll
        for (int r = 0; r < 8; r++) {
            int t = r + thalf;
            log_s[t * 48 + nt * 16 + mrow] = c[r];
        }
    }
    __syncthreads();

    // ---- Phase 3: action softmax over {push,pop,noop}, one thread per (t,head)
    if (tid < 256) {
        int t = tid >> 4, h = tid & 15;
        float l0 = (log_s[t * 48 + h * 3 + 0] + ba[h * 3 + 0]) * 0.125f;
        float l1 = (log_s[t * 48 + h * 3 + 1] + ba[h * 3 + 1]) * 0.125f;
        float l2 = (log_s[t * 48 + h * 3 + 2] + ba[h * 3 + 2]) * 0.125f;
        float mx = fmaxf(l0, fmaxf(l1, l2));
        float e0 = __expf(l0 - mx), e1 = __expf(l1 - mx), e2 = __expf(l2 - mx);
        float inv = 1.0f / (e0 + e1 + e2);
        act_s[(t * 16 + h) * 3 + 0] = e0 * inv;
        act_s[(t * 16 + h) * 3 + 1] = e1 * inv;
        act_s[(t * 16 + h) * 3 + 2] = e2 * inv;
    }
    __syncthreads();

    // ---- Phase 4: fused stack update + gated readout, one wave per head ----
    {
        const int h = wave;
        const float wg  = Wg[lane];   // lane == sd index
        const float bg0 = bg[0];
        for (int t = 0; t < TOK_PER_WG; t++) {
            const long long tok   = tok0 + t;
            const long long sbase = ((tok * NHEAD) + h) * (NSLOTS * SDIM);
            const long long mbase = ((tok * NHEAD) + h) * NSLOTS;
            const float ap   = act_s[(t * 16 + h) * 3 + 0];
            const float apop = act_s[(t * 16 + h) * 3 + 1];
            const float an   = act_s[(t * 16 + h) * 3 + 2];

            float st[NSLOTS];
            #pragma unroll
            for (int i = 0; i < NSLOTS; i++)        // 128B coalesced per slot
                st[i] = stack_g[sbase + i * SDIM + lane];
            const float kv = k_s[(h * 16 + t) * 33 + lane];

            float mval = (lane < 16) ? mask_g[mbase + lane] : 0.0f;
            float mall[NSLOTS];
            #pragma unroll
            for (int i = 0; i < NSLOTS; i++) mall[i] = __shfl(mval, i, 32);

            float ns[NSLOTS], nm[NSLOTS];
            #pragma unroll
            for (int i = 0; i < NSLOTS; i++) {
                float pv = (i == 0)  ? kv   : st[i - 1];
                float qv = (i == 15) ? 0.0f : st[i + 1];
                ns[i] = ap * pv + apop * qv + an * st[i];
                float pm = (i == 0)  ? 1.0f : mall[i - 1];
                float qm = (i == 15) ? 0.0f : mall[i + 1];
                nm[i] = ap * pm + apop * qm + an * mall[i];
            }
            #pragma unroll
            for (int i = 0; i < NSLOTS; i++)
                ostack_g[sbase + i * SDIM + lane] = ns[i];

            // lane-shift form of new_mask for the store (no runtime reg index)
            {
                float pm = __shfl_up(mval, 1, 32);
                if (lane == 0) pm = 1.0f;
                float qm = __shfl_down(mval, 1, 32);   // lane15 reads lane16 (=0)
                float nml = ap * pm + apop * qm + an * mval;
                if (lane < 16) omask_g[mbase + lane] = nml;
            }

            // gate scores: per-slot dot over sd via xor-butterfly reduction
            float sc[NSLOTS];
            #pragma unroll
            for (int i = 0; i < NSLOTS; i++) {
                float p = ns[i] * wg;
                p += __shfl_xor(p, 16, 32);
                p += __shfl_xor(p, 8, 32);
                p += __shfl_xor(p, 4, 32);
                p += __shfl_xor(p, 2, 32);
                p += __shfl_xor(p, 1, 32);
                sc[i] = p + bg0 + (1.0f - nm[i]) * -1e9f;
            }
            float mx = sc[0];
            #pragma unroll
            for (int i = 1; i < NSLOTS; i++) mx = fmaxf(mx, sc[i]);
            float denom = 0.0f, ex[NSLOTS];
            #pragma unroll
            for (int i = 0; i < NSLOTS; i++) { ex[i] = __expf(sc[i] - mx); denom += ex[i]; }
            float inv = 1.0f / denom;
            float memv = 0.0f;
            #pragma unroll
            for (int i = 0; i < NSLOTS; i++) memv += ex[i] * inv * ns[i];
            mem_s[(h * 16 + t) * 33 + lane] = memv;
        }
    }
    __syncthreads();

    // ---- Phase 5: out[16x64] = mem[16x32] @ Wu, + bias, *res, + residual ----
    {
        const int h = wave;
        v8f c[4] = {};
        #pragma unroll
        for (int k0 = 0; k0 < SDIM; k0 += 4) {
            v2f a;
            a.x = mem_s[(h * 16 + mrow) * 33 + k0 + khi];
            a.y = mem_s[(h * 16 + mrow) * 33 + k0 + khi + 1];
            #pragma unroll
            for (int nt = 0; nt < 4; nt++) {
                v2f b;
                b.x = Wu[(k0 + khi) * HDIM + nt * 16 + mrow];
                b.y = Wu[(k0 + khi + 1) * HDIM + nt * 16 + mrow];
                c[nt] = __builtin_amdgcn_wmma_f32_16x16x4_f32(false, a, false, b,
                                                              (short)0, c[nt], false, false);
            }
        }
        const float rw = resw[0];
        #pragma unroll
        for (int nt = 0; nt < 4; nt++) {
            int ncol = nt * 16 + mrow;
            float buv = bu[ncol];
            #pragma unroll
            for (int r = 0; r < 8; r++) {
                int t = r + thalf;
                float v = (c[nt][r] + buv) * rw + hs_s[t * HS_STRIDE + h * HDIM + ncol];
                out_g[(tok0 + t) * HH + h * HDIM + ncol] = v;
            }
        }
    }
}

extern "C" void kernel_launch(void* const* d_in, const int* in_sizes, int n_in,
                              void* d_out, int out_size, void* d_ws, size_t ws_size,
                              hipStream_t stream) {
    const float* hs    = (const float*)d_in[0];
    const float* stack = (const float*)d_in[1];
    const float* mask  = (const float*)d_in[2];
    const float* Wa    = (const float*)d_in[3];
    const float* ba    = (const float*)d_in[4];
    const float* Wd    = (const float*)d_in[5];
    const float* bd    = (const float*)d_in[6];
    const float* Wu    = (const float*)d_in[7];
    const float* bu    = (const float*)d_in[8];
    const float* Wg    = (const float*)d_in[9];
    const float* bg    = (const float*)d_in[10];
    const float* resw  = (const float*)d_in[11];

    float* out    = (float*)d_out;
    float* ostack = out + (size_t)NB * SS * HH;
    float* omask  = ostack + (size_t)NB * SS * NHEAD * NSLOTS * SDIM;

    const size_t smem = SMEM_FLOATS * sizeof(float);   // ~139 KB (< 320 KB/WGP)
    dim3 grid((NB * SS) / TOK_PER_WG);                 // 512 workgroups
    stackmem_kernel<<<grid, THREADS, smem, stream>>>(
        hs, stack, mask, Wa, ba, Wd, bd, Wu, bu, Wg, bg, resw,
        out, ostack, omask);
}